// TripletLossHuman_85272280695081
// MI455X (gfx1250) — compile-verified
//
#include <hip/hip_runtime.h>
#include <hip/hip_bf16.h>
#include <math.h>

#define NUM_CLASSES 16
#define MARGIN_F 0.3f
#define EPS_F 1e-8f
#define NBLK 256

typedef __attribute__((ext_vector_type(2))) float v2f;
typedef __attribute__((ext_vector_type(8))) float v8f;

// ---------------------------------------------------------------------------
// Kernel 1: first occurrence index per class + presence mask.
// targets: B int32 class ids in [0,16). One block, LDS atomicMin.
// ---------------------------------------------------------------------------
__global__ void k_first_idx(const int* __restrict__ targets, int n,
                            int* __restrict__ first_idx,
                            int* __restrict__ present) {
    __shared__ int s_min[NUM_CLASSES];
    const int t = threadIdx.x;
    if (t < NUM_CLASSES) s_min[t] = 0x7fffffff;
    __syncthreads();
    for (int i = t; i < n; i += blockDim.x) {
        int c = targets[i];
        if (c >= 0 && c < NUM_CLASSES) atomicMin(&s_min[c], i);
    }
    __syncthreads();
    if (t < NUM_CLASSES) {
        int m = s_min[t];
        int p = (m != 0x7fffffff) ? 1 : 0;
        present[t]   = p;
        first_idx[t] = p ? m : 0;   // argmax-of-all-false == 0, matches jnp
    }
}

// ---------------------------------------------------------------------------
// Kernel 2: single-wave WMMA Gram matrix G = Xrep * Xrep^T (16x16, K=D in
// steps of 4 via V_WMMA_F32_16X16X4_F32), then dist16[m][n] =
// sqrt(max(G[m][m]+G[n][n]-2 G[m][n], 1e-12)).
//
// f32 16x4 A layout (wave32): lanes 0-15 -> M=lane, VGPR0/1 = K=0/1;
// lanes 16-31 -> M=lane-16, VGPR0/1 = K=2/3. The B (=A^T slice of the same
// X tile) layout coincides, so one float2 load per lane feeds both operands.
// ---------------------------------------------------------------------------
__global__ void k_gram_dist(const float* __restrict__ X, int D,
                            const int* __restrict__ first_idx,
                            float* __restrict__ dist16) {
    const int lane  = threadIdx.x;        // 0..31, EXEC all ones (WMMA req.)
    const int rloc  = lane & 15;          // matrix row this lane covers
    const int khalf = (lane >> 4) << 1;   // 0 or 2: which K pair

    const long row = (long)first_idx[rloc];
    const float2* __restrict__ xr2 =
        (const float2*)(X + row * (long)D) + (khalf >> 1);

    v8f acc = {};
    #pragma unroll 8
    for (int kb = 0; kb < D; kb += 4) {
        float2 f = xr2[kb >> 1];          // X[row][kb+khalf], X[row][kb+khalf+1]
        v2f a; a.x = f.x; a.y = f.y;
        // D = A * B + C ; A = B = same 16x4 f32 tile of Xrep
        acc = __builtin_amdgcn_wmma_f32_16x16x4_f32(
            /*neg_a=*/false, a, /*neg_b=*/false, a,
            /*c_mod=*/(short)0, acc, /*reuse_a=*/false, /*reuse_b=*/false);
    }

    // Unpack D layout through LDS: VGPR r holds M=r (lanes 0-15) / M=r+8.
    __shared__ float G[16][16];
    const int ncol  = lane & 15;
    const int mbase = (lane >> 4) << 3;
    #pragma unroll
    for (int r = 0; r < 8; ++r) G[mbase + r][ncol] = acc[r];
    __syncthreads();

    for (int e = lane; e < 256; e += 32) {
        int m = e >> 4, n = e & 15;
        float d2 = G[m][m] + G[n][n] - 2.0f * G[m][n];
        dist16[e] = sqrtf(fmaxf(d2, 1e-12f));
    }
}

// ---------------------------------------------------------------------------
// Kernel 3: masked triplet reduction over user_answers (T x 3 int32),
// per-block partial sums (rank, perplexity, valid-count).
// ---------------------------------------------------------------------------
__global__ void k_triplet_partial(const int* __restrict__ ans, int T,
                                  const float* __restrict__ dist16,
                                  const int* __restrict__ present,
                                  float* __restrict__ partials) {
    __shared__ float s_dist[256];
    __shared__ int   s_pres[NUM_CLASSES];
    const int t = threadIdx.x;
    s_dist[t] = dist16[t];                 // blockDim == 256
    if (t < NUM_CLASSES) s_pres[t] = present[t];
    __syncthreads();

    float sr = 0.0f, sp = 0.0f, sv = 0.0f;
    for (int i = blockIdx.x * blockDim.x + t; i < T;
         i += gridDim.x * blockDim.x) {
        const int a  = ans[3 * i + 0];
        const int p  = ans[3 * i + 1];
        const int ng = ans[3 * i + 2];
        if (s_pres[a] & s_pres[p] & s_pres[ng]) {
            const float dap = s_dist[a * 16 + p];
            const float dan = s_dist[a * 16 + ng];
            sr += fmaxf(dap - dan + MARGIN_F, 0.0f);
            const float s_ap = 1.0f / (dap + 1.0f);
            const float s_an = 1.0f / (dan + 1.0f);
            sp += -logf(s_ap / (s_ap + s_an) + EPS_F);
            sv += 1.0f;
        }
    }

    __shared__ float red0[256], red1[256], red2[256];
    red0[t] = sr; red1[t] = sp; red2[t] = sv;
    __syncthreads();
    for (int off = 128; off > 0; off >>= 1) {
        if (t < off) {
            red0[t] += red0[t + off];
            red1[t] += red1[t + off];
            red2[t] += red2[t + off];
        }
        __syncthreads();
    }
    if (t == 0) {
        partials[blockIdx.x * 3 + 0] = red0[0];
        partials[blockIdx.x * 3 + 1] = red1[0];
        partials[blockIdx.x * 3 + 2] = red2[0];
    }
}

// ---------------------------------------------------------------------------
// Kernel 4: deterministic fixed-order final reduction + loss assembly.
// out = [loss, loss_human, loss_perplexity] (f32).
// ---------------------------------------------------------------------------
__global__ void k_final(const float* __restrict__ partials,
                        float* __restrict__ out) {
    __shared__ float red0[NBLK], red1[NBLK], red2[NBLK];
    const int t = threadIdx.x;
    red0[t] = partials[t * 3 + 0];
    red1[t] = partials[t * 3 + 1];
    red2[t] = partials[t * 3 + 2];
    __syncthreads();
    for (int off = NBLK / 2; off > 0; off >>= 1) {
        if (t < off) {
            red0[t] += red0[t + off];
            red1[t] += red1[t + off];
            red2[t] += red2[t + off];
        }
        __syncthreads();
    }
    if (t == 0) {
        const float nv = fmaxf(red2[0], 1.0f);
        const float lh = red0[0] / nv;
        const float lp = red1[0] / nv;
        out[0] = lh + lp;   // W_HUMAN * lh + W_PER * lp, weights = 1
        out[1] = lh;
        out[2] = lp;
    }
}

// ---------------------------------------------------------------------------
// Launch. Inputs (setup_inputs order):
// 0: preds_mat (unused)  1: preds_sub (unused)  2: inputs f32 (B x D)
// 3: targets_mat int (B) 4: targets_sub (unused) 5: user_answers int (T x 3)
// ---------------------------------------------------------------------------
extern "C" void kernel_launch(void* const* d_in, const int* in_sizes, int n_in,
                              void* d_out, int out_size, void* d_ws,
                              size_t ws_size, hipStream_t stream) {
    const float* X       = (const float*)d_in[2];
    const int*   targets = (const int*)d_in[3];
    const int*   answers = (const int*)d_in[5];
    float*       out     = (float*)d_out;

    const int B = in_sizes[3];
    const int D = in_sizes[2] / B;       // 1024
    const int T = in_sizes[5] / 3;       // 100000

    // Workspace layout (16-byte friendly):
    //   [0)    int  first_idx[16]
    //   [64)   int  present[16]
    //   [128)  f32  dist16[256]
    //   [1152) f32  partials[NBLK*3]
    char* ws        = (char*)d_ws;
    int*   first_i  = (int*)(ws + 0);
    int*   present  = (int*)(ws + 64);
    float* dist16   = (float*)(ws + 128);
    float* partials = (float*)(ws + 1152);

    k_first_idx<<<1, 256, 0, stream>>>(targets, B, first_i, present);
    k_gram_dist<<<1, 32, 0, stream>>>(X, D, first_i, dist16);
    k_triplet_partial<<<NBLK, 256, 0, stream>>>(answers, T, dist16, present,
                                                partials);
    k_final<<<1, NBLK, 0, stream>>>(partials, out);
}